// GemmRSTritonPCIe_86646670229697
// MI455X (gfx1250) — compile-verified
//
#include <hip/hip_runtime.h>

typedef __attribute__((ext_vector_type(2))) float v2f;
typedef __attribute__((ext_vector_type(8))) float v8f;

// Problem constants (from reference)
constexpr int Wd = 4;
constexpr int Md = 4096;
constexpr int Kd = 2048;
constexpr int Nd = 4096;

// Tiling
constexpr int BM = 128;          // block tile M
constexpr int BN = 128;          // block tile N
constexpr int KB = 32;           // k-slab per LDS stage
constexpr int LDSA = KB + 4;     // padded row stride in floats (144B: 16B aligned, bank-spread)
constexpr int SLABS = Wd * (Kd / KB);   // 256 k-slabs total (w-sum folded into K)

// Async global->LDS 16B copy (GVS mode: SGPR64 base + per-lane u32 byte offset).
// VDST holds the LDS byte address (low 32 bits of flat shared pointer per aperture rule).
__device__ __forceinline__ void async_ld_b128(unsigned ldsAddr, unsigned gByteOff,
                                              const float* __restrict__ base)
{
    asm volatile("global_load_async_to_lds_b128 %0, %1, %2"
                 :: "v"(ldsAddr), "v"(gByteOff), "s"(base)
                 : "memory");
}

// Issue one 128x32-f32 tile pair (A from x, B from weight): 4 slots/thread/tile => 8 async ops.
__device__ __forceinline__ void stage_slab(const float* __restrict__ xw,
                                           const float* __restrict__ ww,
                                           float* AsBuf, float* BsBuf,
                                           int bm, int bn, int k0, int tid)
{
    const unsigned aBase = (unsigned)(uintptr_t)AsBuf;
    const unsigned bBase = (unsigned)(uintptr_t)BsBuf;
    #pragma unroll
    for (int r = 0; r < 4; ++r) {
        const int s   = tid + r * 256;
        const int row = s >> 3;           // 0..127
        const int c4  = (s & 7) << 2;     // 0,4,...,28 floats
        const unsigned ldsOff = (unsigned)((row * LDSA + c4) * 4);
        async_ld_b128(aBase + ldsOff,
                      (unsigned)(((size_t)(bm + row) * Kd + k0 + c4) * 4), xw);
        async_ld_b128(bBase + ldsOff,
                      (unsigned)(((size_t)(bn + row) * Kd + k0 + c4) * 4), ww);
    }
}

__global__ __launch_bounds__(256)
void gemm_rs_wmma_f32(const float* __restrict__ x,
                      const float* __restrict__ w,
                      const float* __restrict__ bias,
                      float* __restrict__ out)
{
    __shared__ float As[2][BM * LDSA];   // x tile:  [m][k], double buffered
    __shared__ float Bs[2][BN * LDSA];   // w tile:  [n][k], double buffered

    const int tid   = threadIdx.x;
    const int lane  = tid & 31;
    const int wave  = tid >> 5;       // 0..7
    const int wm    = wave & 1;       // 2 waves along M -> 64 rows each
    const int wn    = wave >> 1;      // 4 waves along N -> 32 cols each

    const int bm = blockIdx.x * BM;
    const int bn = blockIdx.y * BN;

    const int lhalf = lane >> 4;      // selects k pair per ISA A/B frag layout
    const int l15   = lane & 15;

    v8f acc[4][2] = {};               // 4 M-tiles x 2 N-tiles of 16x16 f32

    // Prologue: stage slab 0 into buffer 0.
    stage_slab(x, w, As[0], Bs[0], bm, bn, 0, tid);

    for (int s = 0; s < SLABS; ++s) {
        const int buf = s & 1;

        // All waves finished reading buf^1 (previous compute) before we async-write it.
        __syncthreads();

        if (s + 1 < SLABS) {
            const int ns  = s + 1;
            const int wi  = ns >> 6;              // Kd/KB == 64 slabs per rank
            const int k0  = (ns & 63) * KB;
            stage_slab(x + (size_t)wi * Md * Kd,
                       w + (size_t)wi * Nd * Kd,
                       As[buf ^ 1], Bs[buf ^ 1], bm, bn, k0, tid);
            // 8 newer ops in flight; in-order completion => slab s fully landed in LDS.
            asm volatile("s_wait_asynccnt 0x8" ::: "memory");
        } else {
            asm volatile("s_wait_asynccnt 0x0" ::: "memory");
        }

        // Slab s visible to every wave.
        __syncthreads();

        const float* Asb = As[buf];
        const float* Bsb = Bs[buf];

        #pragma unroll
        for (int kk = 0; kk < KB; kk += 4) {
            v2f afrag[4];
            v2f bfrag[2];
            #pragma unroll
            for (int mt = 0; mt < 4; ++mt) {
                const int row = wm * 64 + mt * 16 + l15;
                afrag[mt] = *(const v2f*)(Asb + row * LDSA + kk + 2 * lhalf);
            }
            #pragma unroll
            for (int nt = 0; nt < 2; ++nt) {
                const int row = wn * 32 + nt * 16 + l15;
                bfrag[nt] = *(const v2f*)(Bsb + row * LDSA + kk + 2 * lhalf);
            }
            #pragma unroll
            for (int mt = 0; mt < 4; ++mt) {
                #pragma unroll
                for (int nt = 0; nt < 2; ++nt) {
                    acc[mt][nt] = __builtin_amdgcn_wmma_f32_16x16x4_f32(
                        /*neg_a=*/false, afrag[mt],
                        /*neg_b=*/false, bfrag[nt],
                        /*c_mod=*/(short)0, acc[mt][nt],
                        /*reuse_a=*/false, /*reuse_b=*/false);
                }
            }
        }
    }

    // Epilogue: out[m,n] = acc + sum_w bias[w,m,n].  C/D layout: VGPR r, lanes0-15 -> M=r,
    // lanes16-31 -> M=r+8; N = l15.  Stores coalesced over lanes.
    const size_t planeSz = (size_t)Md * Nd;
    #pragma unroll
    for (int mt = 0; mt < 4; ++mt) {
        #pragma unroll
        for (int nt = 0; nt < 2; ++nt) {
            #pragma unroll
            for (int r = 0; r < 8; ++r) {
                const int m = bm + wm * 64 + mt * 16 + r + 8 * lhalf;
                const int n = bn + wn * 32 + nt * 16 + l15;
                const size_t off = (size_t)m * Nd + n;
                float b = bias[off]
                        + bias[off +     planeSz]
                        + bias[off + 2 * planeSz]
                        + bias[off + 3 * planeSz];
                out[off] = acc[mt][nt][r] + b;
            }
        }
    }
}

extern "C" void kernel_launch(void* const* d_in, const int* in_sizes, int n_in,
                              void* d_out, int out_size, void* d_ws, size_t ws_size,
                              hipStream_t stream)
{
    (void)in_sizes; (void)n_in; (void)out_size; (void)d_ws; (void)ws_size;
    const float* x    = (const float*)d_in[0];   // [W, M, K]
    const float* wgt  = (const float*)d_in[1];   // [W, N, K]
    const float* bias = (const float*)d_in[2];   // [W, M, N]
    float* out        = (float*)d_out;           // [M, N] (== [W, M/W, N] flat)

    dim3 grid(Md / BM, Nd / BN, 1);              // 32 x 32 blocks
    dim3 block(256, 1, 1);                       // 8 waves (wave32)
    gemm_rs_wmma_f32<<<grid, block, 0, stream>>>(x, wgt, bias, out);
}